// MultiBoxLoss_81570018886370
// MI455X (gfx1250) — compile-verified
//
#include <hip/hip_runtime.h>
#include <math.h>

// Problem constants (match reference setup_inputs)
static constexpr int kB = 64;      // batch
static constexpr int kP = 24564;   // priors
static constexpr int kO = 16;      // objects per image
static constexpr int kC = 21;      // classes
static constexpr float kThr = 0.5f;
static constexpr float kVar0 = 0.1f;
static constexpr float kVar1 = 0.2f;

typedef __attribute__((ext_vector_type(2))) float v2f;
typedef __attribute__((ext_vector_type(8))) float v8f;

#if defined(__gfx1250__) && __has_builtin(__builtin_amdgcn_wmma_f32_16x16x4_f32)
#define USE_WMMA_F32_16X16X4 1
#endif

// ---------------------------------------------------------------------------
// K0: zero accumulators / argmax slots (ws is poisoned by the harness)
// ---------------------------------------------------------------------------
__global__ void k_init(unsigned long long* __restrict__ slot,
                       int* __restrict__ numPos, unsigned* __restrict__ selT,
                       int* __restrict__ selK, float* __restrict__ lossAcc) {
  int i = blockIdx.x * blockDim.x + threadIdx.x;
  if (i < kB * kO) slot[i] = 0ull;
  if (i < kB) { numPos[i] = 0; selT[i] = 0u; selK[i] = 0; }
  if (i < 2) lossAcc[i] = 0.0f;
}

// ---------------------------------------------------------------------------
// K1: per-(b,p) best truth (max/argmax over 16 truths), and per-(b,o) best
// prior via block reduction + packed u64 atomicMax (value<<32 | ~p so that
// ties pick the smallest prior index, matching jnp.argmax first-occurrence).
// ---------------------------------------------------------------------------
__global__ void k_match(const float* __restrict__ priors,
                        const float* __restrict__ targets,
                        float* __restrict__ bestOv, int* __restrict__ bestIdx,
                        unsigned long long* __restrict__ slot) {
  const int b = blockIdx.y;
  const int tid = threadIdx.x;
  const int p = blockIdx.x * 256 + tid;
  const bool valid = p < kP;
  const int pc = valid ? p : 0;

  const float cx = priors[pc * 4 + 0], cy = priors[pc * 4 + 1];
  const float w  = priors[pc * 4 + 2], h  = priors[pc * 4 + 3];
  const float px0 = cx - 0.5f * w, py0 = cy - 0.5f * h;
  const float px1 = cx + 0.5f * w, py1 = cy + 0.5f * h;
  const float areaP = w * h;

  float bestV = -1.0f;
  int bestO = 0;
  __shared__ unsigned long long s64[256];

  for (int o = 0; o < kO; ++o) {
    const size_t tb = ((size_t)b * kO + o) * 5;
    const float t0 = targets[tb + 0], t1 = targets[tb + 1];
    const float t2 = targets[tb + 2], t3 = targets[tb + 3];
    float iw = fminf(px1, t2) - fmaxf(px0, t0); iw = fmaxf(iw, 0.0f);
    float ih = fminf(py1, t3) - fmaxf(py0, t1); ih = fmaxf(ih, 0.0f);
    const float inter = iw * ih;
    const float areaT = (t2 - t0) * (t3 - t1);
    const float iou = inter / (areaT + areaP - inter);
    if (valid && iou > bestV) { bestV = iou; bestO = o; }

    unsigned long long pk = 0ull;
    if (valid) {
      pk = (((unsigned long long)__float_as_uint(iou)) << 32) |
           (unsigned long long)(0xFFFFFFFFu - (unsigned)p);
    }
    s64[tid] = pk;
    __syncthreads();
    for (int s = 128; s > 0; s >>= 1) {
      if (tid < s) {
        unsigned long long o2 = s64[tid + s];
        if (o2 > s64[tid]) s64[tid] = o2;
      }
      __syncthreads();
    }
    if (tid == 0) atomicMax(&slot[b * kO + o], s64[0]);
    __syncthreads();
  }
  if (valid) {
    bestOv[(size_t)b * kP + p] = bestV;
    bestIdx[(size_t)b * kP + p] = bestO;
  }
}

// ---------------------------------------------------------------------------
// K2: sequential override per batch (last truth wins, like the scatter)
// ---------------------------------------------------------------------------
__global__ void k_override(const unsigned long long* __restrict__ slot,
                           float* __restrict__ bestOv, int* __restrict__ bestIdx) {
  const int b = blockIdx.x * blockDim.x + threadIdx.x;
  if (b >= kB) return;
  for (int o = 0; o < kO; ++o) {
    const unsigned long long s = slot[b * kO + o];
    const unsigned p = 0xFFFFFFFFu - (unsigned)(s & 0xFFFFFFFFull);
    bestOv[(size_t)b * kP + p] = 2.0f;
    bestIdx[(size_t)b * kP + p] = o;
  }
}

// ---------------------------------------------------------------------------
// K3: fused logsumexp (sum-of-exp via V_WMMA_F32_16X16X4_F32 with B==ones),
// cross-entropy, loc encode + smooth-L1, pos count.
// Each wave handles 16 priors: lanes L and L+16 split prior (L&15)'s classes
// exactly along the WMMA 16x4 A-matrix layout (K=0,1 vs K=2,3 per chunk).
// ---------------------------------------------------------------------------
__global__ void k_ce(const float* __restrict__ loc, const float* __restrict__ conf,
                     const float* __restrict__ priors, const float* __restrict__ targets,
                     const float* __restrict__ bestOv, const int* __restrict__ bestIdx,
                     float* __restrict__ ceArr, float* __restrict__ lossHard,
                     int* __restrict__ confT, int* __restrict__ numPos,
                     float* __restrict__ lossAcc) {
  const int b = blockIdx.y;
  const int tid = threadIdx.x;
  const int wave = tid >> 5;
  const int lane = tid & 31;
  const int pr16 = lane & 15;
  const bool lower = lane < 16;

  const int p = blockIdx.x * 128 + wave * 16 + pr16;
  const bool valid = p < kP;
  const int pc = valid ? p : (kP - 1);
  const size_t row = ((size_t)b * kP + pc) * kC;

  // Each lane loads its interleaved class pairs: lower lanes {4k,4k+1},
  // upper lanes {4k+2,4k+3}; pads (>=21) contribute -inf to max / 0 to exp.
  float v0[6], v1[6];
  float m = -1e30f;
#pragma unroll
  for (int kk = 0; kk < 6; ++kk) {
    const int c0 = 4 * kk + (lower ? 0 : 2);
    const int c1 = c0 + 1;
    const float a0 = (c0 < kC) ? conf[row + c0] : -1e30f;
    const float a1 = (c1 < kC) ? conf[row + c1] : -1e30f;
    v0[kk] = a0; v1[kk] = a1;
    m = fmaxf(m, fmaxf(a0, a1));
  }
  m = fmaxf(m, __shfl_xor(m, 16, 32));  // full 21-class max for this prior

  float sumexp;
#ifdef USE_WMMA_F32_16X16X4
  {
    v8f acc = {};
    const v2f ones = {1.0f, 1.0f};
#pragma unroll
    for (int kk = 0; kk < 6; ++kk) {
      const int c0 = 4 * kk + (lower ? 0 : 2);
      const int c1 = c0 + 1;
      v2f a = {(c0 < kC) ? expf(v0[kk] - m) : 0.0f,
               (c1 < kC) ? expf(v1[kk] - m) : 0.0f};
      // D[mrow][n] += sum_k A[mrow][k] * 1  -> per-prior sum replicated over N
      acc = __builtin_amdgcn_wmma_f32_16x16x4_f32(
          /*neg_a=*/false, a, /*neg_b=*/false, ones,
          /*c_mod=*/(short)0, acc, /*reuse_a=*/false, /*reuse_b=*/false);
    }
    // D layout: lanes 0-15 hold rows M=0..7 in acc[0..7]; lanes 16-31 rows 8..15.
    const int ip = pr16 & 7;
    const float vl =
        (ip & 4) ? ((ip & 2) ? ((ip & 1) ? acc[7] : acc[6])
                             : ((ip & 1) ? acc[5] : acc[4]))
                 : ((ip & 2) ? ((ip & 1) ? acc[3] : acc[2])
                             : ((ip & 1) ? acc[1] : acc[0]));
    const int srcLane = (pr16 < 8) ? pr16 : (pr16 + 8);
    sumexp = __shfl(vl, srcLane, 32);
  }
#else
  {
    float s = 0.0f;
#pragma unroll
    for (int kk = 0; kk < 6; ++kk) {
      const int c0 = 4 * kk + (lower ? 0 : 2);
      const int c1 = c0 + 1;
      s += (c0 < kC) ? expf(v0[kk] - m) : 0.0f;
      s += (c1 < kC) ? expf(v1[kk] - m) : 0.0f;
    }
    sumexp = s + __shfl_xor(s, 16, 32);
  }
#endif
  const float lse = m + logf(sumexp);

  float locSum = 0.0f;
  int posCnt = 0;
  if (lower && valid) {
    const size_t gidx = (size_t)b * kP + p;
    const float bo = bestOv[gidx];
    const int bi = bestIdx[gidx];
    const float lab = targets[((size_t)b * kO + bi) * 5 + 4];
    int ct = (int)lab + 1;
    if (bo < kThr) ct = 0;
    const float gathered = conf[row + ct];
    const float cei = lse - gathered;
    ceArr[gidx] = cei;
    confT[gidx] = ct;
    lossHard[gidx] = (ct > 0) ? 0.0f : cei;
    if (ct > 0) {
      posCnt = 1;
      // encode
      const float mx0 = targets[((size_t)b * kO + bi) * 5 + 0];
      const float my0 = targets[((size_t)b * kO + bi) * 5 + 1];
      const float mx1 = targets[((size_t)b * kO + bi) * 5 + 2];
      const float my1 = targets[((size_t)b * kO + bi) * 5 + 3];
      const float cx = priors[p * 4 + 0], cy = priors[p * 4 + 1];
      const float w  = priors[p * 4 + 2], h  = priors[p * 4 + 3];
      const float g0 = ((mx0 + mx1) * 0.5f - cx) / (kVar0 * w);
      const float g1 = ((my0 + my1) * 0.5f - cy) / (kVar0 * h);
      const float g2 = logf((mx1 - mx0) / w) / kVar1;
      const float g3 = logf((my1 - my0) / h) / kVar1;
      const size_t lb = gidx * 4;
      const float g[4] = {g0, g1, g2, g3};
#pragma unroll
      for (int j = 0; j < 4; ++j) {
        const float d = loc[lb + j] - g[j];
        const float ad = fabsf(d);
        locSum += (ad < 1.0f) ? 0.5f * d * d : (ad - 0.5f);
      }
    }
  }

  __shared__ float sf[256];
  __shared__ int si[256];
  sf[tid] = locSum; si[tid] = posCnt;
  __syncthreads();
  for (int s = 128; s > 0; s >>= 1) {
    if (tid < s) { sf[tid] += sf[tid + s]; si[tid] += si[tid + s]; }
    __syncthreads();
  }
  if (tid == 0) {
    atomicAdd(&lossAcc[0], sf[0]);
    atomicAdd(&numPos[b], si[0]);
  }
}

// ---------------------------------------------------------------------------
// K4: per-batch radix select (MSB-first, 4x8-bit) to find the num_neg-th
// largest loss_hard value T and the remaining tie count k_rem.
// ---------------------------------------------------------------------------
__global__ void k_select(const float* __restrict__ lossHard,
                         const int* __restrict__ numPos,
                         unsigned* __restrict__ selT, int* __restrict__ selK) {
  const int b = blockIdx.x;
  const int tid = threadIdx.x;
  __shared__ unsigned hist[256];
  __shared__ unsigned s_prefix;
  __shared__ int s_kk;
  __shared__ int s_empty;

  if (tid == 0) {
    const int np = numPos[b];
    int k = 3 * np;
    if (k > kP - 1) k = kP - 1;
    s_prefix = 0u; s_kk = k; s_empty = (k <= 0);
  }
  __syncthreads();
  if (s_empty) {
    if (tid == 0) { selT[b] = 0xFFFFFFFFu; selK[b] = 0; }
    return;
  }
  for (int pass = 0; pass < 4; ++pass) {
    const int shift = 24 - 8 * pass;
    const unsigned mask = (pass == 0) ? 0u : (0xFFFFFFFFu << (shift + 8));
    hist[tid] = 0u;
    __syncthreads();
    const unsigned pref = s_prefix;
    for (int i = tid; i < kP; i += 256) {
      const unsigned key = __float_as_uint(lossHard[(size_t)b * kP + i]);
      if ((key & mask) == pref) atomicAdd(&hist[(key >> shift) & 255], 1u);
    }
    __syncthreads();
    if (tid == 0) {
      int kk = s_kk;
      int bin = 255;
      for (; bin > 0; --bin) {
        const int c = (int)hist[bin];
        if (kk <= c) break;
        kk -= c;
      }
      s_kk = kk;
      s_prefix = pref | ((unsigned)bin << shift);
    }
    __syncthreads();
  }
  if (tid == 0) { selT[b] = s_prefix; selK[b] = s_kk; }
}

// ---------------------------------------------------------------------------
// K5: per-batch ordered pass. neg iff key>T, or key==T and its stable-order
// tie rank (over ALL elements, incl. positives) < k_rem. loss_c sums ce over
// pos | neg. Exclusive block scan + running base gives the tie ranks.
// ---------------------------------------------------------------------------
__global__ void k_lossc(const float* __restrict__ ceArr,
                        const float* __restrict__ lossHard,
                        const int* __restrict__ confT,
                        const unsigned* __restrict__ selT,
                        const int* __restrict__ selK,
                        float* __restrict__ lossAcc) {
  const int b = blockIdx.x;
  const int tid = threadIdx.x;
  __shared__ int s_scan[256];
  __shared__ float s_red[256];
  const unsigned T = selT[b];
  const int krem = selK[b];
  float local = 0.0f;
  int base = 0;

  for (int start = 0; start < kP; start += 256) {
    const int i = start + tid;
    const bool v = i < kP;
    const size_t idx = (size_t)b * kP + (v ? i : 0);
    const unsigned key = v ? __float_as_uint(lossHard[idx]) : 0u;
    const float cei = v ? ceArr[idx] : 0.0f;
    const int ct = v ? confT[idx] : 0;
    const int eq = (v && key == T) ? 1 : 0;

    s_scan[tid] = eq;
    __syncthreads();
    for (int off = 1; off < 256; off <<= 1) {
      const int val = s_scan[tid];
      const int add = (tid >= off) ? s_scan[tid - off] : 0;
      __syncthreads();
      s_scan[tid] = val + add;
      __syncthreads();
    }
    const int incl = s_scan[tid];
    const int total = s_scan[255];
    const int tieBefore = base + incl - eq;
    const bool pos = v && (ct > 0);
    const bool sel = v && ((key > T) || (eq && tieBefore < krem));
    if (pos || sel) local += cei;
    base += total;
    __syncthreads();
  }

  s_red[tid] = local;
  __syncthreads();
  for (int s = 128; s > 0; s >>= 1) {
    if (tid < s) s_red[tid] += s_red[tid + s];
    __syncthreads();
  }
  if (tid == 0) atomicAdd(&lossAcc[1], s_red[0]);
}

// ---------------------------------------------------------------------------
// K6: finalize
// ---------------------------------------------------------------------------
__global__ void k_final(const int* __restrict__ numPos,
                        const float* __restrict__ lossAcc,
                        float* __restrict__ out) {
  if (threadIdx.x == 0 && blockIdx.x == 0) {
    int s = 0;
    for (int b = 0; b < kB; ++b) s += numPos[b];
    const float N = (float)((s > 1) ? s : 1);
    out[0] = lossAcc[0] / N;
    out[1] = lossAcc[1] / N;
  }
}

// ---------------------------------------------------------------------------
extern "C" void kernel_launch(void* const* d_in, const int* in_sizes, int n_in,
                              void* d_out, int out_size, void* d_ws, size_t ws_size,
                              hipStream_t stream) {
  const float* loc     = (const float*)d_in[0];  // (B,P,4)
  const float* conf    = (const float*)d_in[1];  // (B,P,21)
  const float* priors  = (const float*)d_in[2];  // (P,4)
  const float* targets = (const float*)d_in[3];  // (B,O,5)
  float* out = (float*)d_out;                    // 2 floats

  char* ws = (char*)d_ws;
  size_t off = 0;
  unsigned long long* slot = (unsigned long long*)(ws + off); off += (size_t)kB * kO * 8;
  float* bestOv   = (float*)(ws + off); off += (size_t)kB * kP * 4;
  int*   bestIdx  = (int*)  (ws + off); off += (size_t)kB * kP * 4;
  float* ceArr    = (float*)(ws + off); off += (size_t)kB * kP * 4;
  float* lossHard = (float*)(ws + off); off += (size_t)kB * kP * 4;
  int*   confT    = (int*)  (ws + off); off += (size_t)kB * kP * 4;
  int*      numPos = (int*)     (ws + off); off += kB * 4;
  unsigned* selT   = (unsigned*)(ws + off); off += kB * 4;
  int*      selK   = (int*)     (ws + off); off += kB * 4;
  float*    lossAcc = (float*)  (ws + off); off += 2 * 4;

  k_init<<<(kB * kO + 255) / 256, 256, 0, stream>>>(slot, numPos, selT, selK, lossAcc);
  k_match<<<dim3((kP + 255) / 256, kB), 256, 0, stream>>>(priors, targets, bestOv, bestIdx, slot);
  k_override<<<1, kB, 0, stream>>>(slot, bestOv, bestIdx);
  k_ce<<<dim3((kP + 127) / 128, kB), 256, 0, stream>>>(loc, conf, priors, targets,
                                                       bestOv, bestIdx, ceArr, lossHard,
                                                       confT, numPos, lossAcc);
  k_select<<<kB, 256, 0, stream>>>(lossHard, numPos, selT, selK);
  k_lossc<<<kB, 256, 0, stream>>>(ceArr, lossHard, confT, selT, selK, lossAcc);
  k_final<<<1, 1, 0, stream>>>(numPos, lossAcc, out);
}